// OneDObjectDetectionLoss_44126493999314
// MI455X (gfx1250) — compile-verified
//
#include <hip/hip_runtime.h>
#include <math.h>
#include <stdint.h>

#define B_ 512
#define A_ 16384
#define C_ 5

typedef __attribute__((ext_vector_type(2))) float v2f;
typedef __attribute__((ext_vector_type(8))) float v8f;

// ---------------------------------------------------------------------------
// Zero the double accumulator in workspace (harness poisons ws each run).
// ---------------------------------------------------------------------------
__global__ void init_acc_kernel(double* acc) {
    if (threadIdx.x == 0 && blockIdx.x == 0) *acc = 0.0;
}

// ---------------------------------------------------------------------------
// Kernel 1: per-batch IoU argmax + regression loss + hot-logit subtraction.
// One block (256 threads) per batch; anchors (128KB) stay hot in L2.
// Tie-break identical to jnp.argmax (first index with the max) via packed
// u64 key: (iou_bits << 32) | (0xFFFFFFFF - idx). IoU >= 0 so float bit
// pattern is monotone; larger ~idx == smaller idx wins among equals.
// ---------------------------------------------------------------------------
__global__ __launch_bounds__(256) void argmax_reg_kernel(
    const float* __restrict__ scores,
    const float* __restrict__ bboxes,
    const int*   __restrict__ gt_classes,
    const float* __restrict__ gt_bboxes,
    const float* __restrict__ anchors,
    double* __restrict__ acc)
{
    __shared__ unsigned long long s_best;
    const int b   = blockIdx.x;
    const int tid = threadIdx.x;
    if (tid == 0) s_best = 0ull;
    __syncthreads();

    const float g0 = gt_bboxes[2 * b];
    const float g1 = gt_bboxes[2 * b + 1];
    const float gw = g1 - g0;
    const float2* anch = (const float2*)anchors;

    float best_iou = -1.0f;
    int   best_idx = 0;
    for (int a = tid; a < A_; a += 256) {
        float2 aa   = anch[a];
        float is    = fmaxf(aa.x, g0);
        float ie    = fminf(aa.y, g1);
        float inter = fmaxf(ie - is, 0.0f);
        float uni   = (aa.y - aa.x) + gw - inter;
        float iou   = inter / uni;
        if (iou > best_iou) { best_iou = iou; best_idx = a; }  // strict > keeps first max
    }

    unsigned long long key =
        ((unsigned long long)__float_as_uint(best_iou) << 32) |
        (unsigned long long)(0xFFFFFFFFu - (unsigned)best_idx);
    atomicMax(&s_best, key);
    __syncthreads();

    if (tid == 0) {
        int best = (int)(0xFFFFFFFFu - (unsigned)(s_best & 0xFFFFFFFFull));
        int cls  = gt_classes[b];
        float x  = scores[((size_t)b * A_ + best) * C_ + cls];
        float2 pb = ((const float2*)bboxes)[(size_t)b * A_ + best];
        float d0 = pb.x - g0, d1 = pb.y - g1;
        float a0 = fabsf(d0), a1 = fabsf(d1);
        float r0 = (a0 < 1.0f) ? 0.5f * d0 * d0 : a0 - 0.5f;
        float r1 = (a1 < 1.0f) ? 0.5f * d1 * d1 : a1 - 0.5f;
        atomicAdd(acc, (double)(r0 + r1 - x));
    }
}

// ---------------------------------------------------------------------------
// Kernel 2: streaming softplus sum over all scores (memory-bound, float4).
// Wave-level reduction done by V_WMMA_F32_16X16X4_F32: lane partials form
// the 16x4 A matrix (vgpr0 = partial, vgpr1 = 0), B = all-ones 4x16, so
// every D element is a row sum p_m + p_{m+16}. Lanes 0-15 hold M=0..7 in
// their 8 D vgprs, lanes 16-31 hold M=8..15; one xor-16 shuffle completes
// the 32-lane sum.
// ---------------------------------------------------------------------------
__device__ __forceinline__ float softplus_term(float x) {
    return fmaxf(x, 0.0f) + log1pf(expf(-fabsf(x)));
}

__global__ __launch_bounds__(256) void softplus_sum_kernel(
    const float4* __restrict__ x4, size_t n4, double* __restrict__ acc)
{
    size_t gid    = (size_t)blockIdx.x * blockDim.x + threadIdx.x;
    size_t stride = (size_t)gridDim.x * blockDim.x;

    float p = 0.0f;
    for (size_t i = gid; i < n4; i += stride) {
        float4 v = x4[i];
        p += softplus_term(v.x);
        p += softplus_term(v.y);
        p += softplus_term(v.z);
        p += softplus_term(v.w);
    }

    // WMMA cross-lane reduction (EXEC is all-ones here: uniform trip counts,
    // loop has fully reconverged).
    v2f a;  a.x = p;    a.y = 0.0f;
    v2f bm; bm.x = 1.0f; bm.y = 1.0f;
    v8f c = {};
    v8f d = __builtin_amdgcn_wmma_f32_16x16x4_f32(
        /*neg_a=*/false, a, /*neg_b=*/false, bm,
        /*c_mod=*/(short)0, c, /*reuse_a=*/false, /*reuse_b=*/false);

    float s = d[0] + d[1] + d[2] + d[3] + d[4] + d[5] + d[6] + d[7];
    float wave_sum = s + __shfl_xor(s, 16, 32);  // lanes 0..15 have M=0..7, 16..31 have M=8..15

    __shared__ float s_w[8];
    int lane = threadIdx.x & 31;
    int wid  = threadIdx.x >> 5;
    if (lane == 0) s_w[wid] = wave_sum;
    __syncthreads();
    if (threadIdx.x == 0) {
        float t = 0.0f;
        #pragma unroll
        for (int i = 0; i < 8; ++i) t += s_w[i];
        atomicAdd(acc, (double)t);
    }
}

// ---------------------------------------------------------------------------
// Finalize: scalar f32 output.
// ---------------------------------------------------------------------------
__global__ void finalize_kernel(const double* __restrict__ acc, float* __restrict__ out) {
    if (threadIdx.x == 0 && blockIdx.x == 0) out[0] = (float)(*acc);
}

extern "C" void kernel_launch(void* const* d_in, const int* in_sizes, int n_in,
                              void* d_out, int out_size, void* d_ws, size_t ws_size,
                              hipStream_t stream) {
    const float* scores     = (const float*)d_in[0];
    const float* bboxes     = (const float*)d_in[1];
    const int*   gt_classes = (const int*)d_in[2];
    const float* gt_bboxes  = (const float*)d_in[3];
    const float* anchors    = (const float*)d_in[4];
    float*  out = (float*)d_out;
    double* acc = (double*)d_ws;

    init_acc_kernel<<<1, 1, 0, stream>>>(acc);

    argmax_reg_kernel<<<B_, 256, 0, stream>>>(scores, bboxes, gt_classes,
                                              gt_bboxes, anchors, acc);

    const size_t n4 = (size_t)B_ * A_ * C_ / 4;  // 10,485,760 float4s
    softplus_sum_kernel<<<8192, 256, 0, stream>>>((const float4*)scores, n4, acc);

    finalize_kernel<<<1, 1, 0, stream>>>(acc, out);
}